// GaussianModel_21534966022500
// MI455X (gfx1250) — compile-verified
//
#include <hip/hip_runtime.h>

// GaussianModel covariance: cov = (R * diag(exp(s))) * (R * diag(exp(s)))^T
// Memory-bound streamer (64 B/item, 512 MB total @ 23.3 TB/s ~ 22 us).
// Gram matrices computed with full-precision V_WMMA_F32_16X16X4_F32:
// 5 items packed per WMMA (rows 0..14 of the 16x4 A operand). The A and B
// operand layouts for this shape are mirror images (lane=M/N, vgpr=K%2,
// lane-half=K/2), so passing the same registers for A and B yields D = A*A^T.
// The 3x3 diagonal blocks of D are the covariances.
//
// D extraction uses the dump-tile-to-LDS trick (2x ds_store_b128 per lane,
// conflict-free padded layout) instead of predicated per-register scatter,
// keeping the dynamic instruction count at/below the HBM roofline.

typedef __attribute__((ext_vector_type(2))) float v2f;
typedef __attribute__((ext_vector_type(4))) float f4;
typedef __attribute__((ext_vector_type(8))) float v8f;

#define IPB   240   // items per block (5 items per WMMA * 48 WMMAs)
#define TPB   256   // 8 wave32
#define NWMMA 48    // IPB / 5
#define CPAD  20    // padded column stride (floats) -> conflict-free b128 dumps

__global__ __launch_bounds__(TPB) void gauss_cov_kernel(
    const float* __restrict__ scaling_raw,   // (N,3) f32
    const float* __restrict__ rotation_raw,  // (N,4) f32
    float* __restrict__ out,                 // (N,3,3) f32
    int N)
{
    // A-operand staging: [wmma][row(M)][K]; K slot 3 zeroed, row 15 zeroed.
    __shared__ float sRS[NWMMA][16][4];          // 12 KB
    // Per-wave D-tile dump, column-major [col N][row M], padded stride.
    __shared__ float dTile[TPB / 32][16][CPAD];  // 10 KB

    const int t    = threadIdx.x;
    const int base = blockIdx.x * IPB;           // base*9 < 2^31, int is safe

    // ---------------- Phase 1: per-item math -> RS rows into LDS ----------
    if (t < IPB) {
        const int i = base + t;
        float w = 0.f, x = 0.f, y = 0.f, z = 0.f;
        float s0 = 0.f, s1 = 0.f, s2 = 0.f;
        if (i < N) {
            const f4 q4 = *(const f4*)(rotation_raw + (size_t)i * 4); // b128
            w = q4.x; x = q4.y; y = q4.z; z = q4.w;
            const float* sp = scaling_raw + (size_t)i * 3;
            s0 = sp[0]; s1 = sp[1]; s2 = sp[2];
        }
        const float sx = expf(s0), sy = expf(s1), sz = expf(s2);

        const float n2  = w*w + x*x + y*y + z*z;
        const float inv = 1.0f / fmaxf(sqrtf(n2), 1e-12f);
        w *= inv; x *= inv; y *= inv; z *= inv;

        const float xx = x*x, yy = y*y, zz = z*z;
        const float xy = x*y, xz = x*z, yz = y*z;
        const float wx = w*x, wy = w*y, wz = w*z;

        const float R00 = 1.f - 2.f*(yy + zz), R01 = 2.f*(xy - wz), R02 = 2.f*(xz + wy);
        const float R10 = 2.f*(xy + wz), R11 = 1.f - 2.f*(xx + zz), R12 = 2.f*(yz - wx);
        const float R20 = 2.f*(xz - wy), R21 = 2.f*(yz + wx), R22 = 1.f - 2.f*(xx + yy);

        const int wi = t / 5;
        const int rb = (t - wi * 5) * 3;         // row base within the 16x4 tile
        float* r0 = &sRS[wi][rb + 0][0];
        r0[0] = R00 * sx; r0[1] = R01 * sy; r0[2] = R02 * sz; r0[3] = 0.f;
        float* r1 = &sRS[wi][rb + 1][0];
        r1[0] = R10 * sx; r1[1] = R11 * sy; r1[2] = R12 * sz; r1[3] = 0.f;
        float* r2 = &sRS[wi][rb + 2][0];
        r2[0] = R20 * sx; r2[1] = R21 * sy; r2[2] = R22 * sz; r2[3] = 0.f;
    }
    if (t < NWMMA) {                             // zero unused row 15 of each tile
        sRS[t][15][0] = 0.f; sRS[t][15][1] = 0.f;
        sRS[t][15][2] = 0.f; sRS[t][15][3] = 0.f;
    }
    __syncthreads();

    // ---------------- Phase 2: 6 WMMAs per wave, D = A * A^T --------------
    const int  wave = t >> 5;
    const int  lane = t & 31;
    const int  Lrow = lane & 15;                 // M (for A) / N (for D)
    const bool hi   = lane >= 16;                // K-half for A, row-half for D

    // u-invariant gather parameters (lanes 0..14, low half, gather one item
    // column each: cov_g[r][cc] = D[rlo+r][Lrow] with Lrow = rlo+cc).
    const int  ii   = Lrow / 3;                  // item index within a WMMA
    const int  cc   = Lrow - 3 * ii;             // column within the item
    const int  rlo  = 3 * ii;
    const bool gat  = (!hi) && (Lrow < 15);
    const int  lim  = N - base;                  // #valid items in this block

    float (*dt)[CPAD] = dTile[wave];
    const float* aptr = &sRS[wave * 6][Lrow][hi ? 2 : 0];
    float* colp = &dt[Lrow][hi ? 8 : 0];

    #pragma unroll
    for (int u = 0; u < 6; ++u) {
        const int wi = wave * 6 + u;
        // A operand: lane<16 -> (K0,K1), lane>=16 -> (K2,K3=0); ds_load_b64.
        const v2f a = *(const v2f*)(aptr + u * 64);   // 64 floats per A tile
        v8f c = {};
        c = __builtin_amdgcn_wmma_f32_16x16x4_f32(
                /*neg_a=*/false, a, /*neg_b=*/false, a,
                /*c_mod=*/(short)0, c, /*reuse_a=*/false, /*reuse_b=*/false);

        // Dump this lane's full D column-half: 2x ds_store_b128, no branches.
        const f4 lo4 = { c[0], c[1], c[2], c[3] };
        const f4 hi4 = { c[4], c[5], c[6], c[7] };
        *(f4*)(colp)     = lo4;
        *(f4*)(colp + 4) = hi4;
        asm volatile("s_wait_dscnt 0" ::: "memory");  // same-wave cross-lane RAW

        // Gather the item's covariance column and store straight to global.
        // Lanes 0..14 collectively cover the WMMA group's contiguous 180 B.
        const int g = wi * 5 + ii;
        if (gat && g < lim) {
            const float* src = &dt[Lrow][rlo];        // rows rlo..rlo+2
            const float c0 = src[0], c1 = src[1], c2 = src[2];
            float* dstp = out + (size_t)(base + g) * 9 + cc;
            dstp[0] = c0; dstp[3] = c1; dstp[6] = c2;
        }
        // Next iteration's dumps cannot pass this iteration's loads:
        // DS ops from the same wave execute in order (WAR safe).
    }
}

extern "C" void kernel_launch(void* const* d_in, const int* in_sizes, int n_in,
                              void* d_out, int out_size, void* d_ws, size_t ws_size,
                              hipStream_t stream) {
    const float* scaling  = (const float*)d_in[0];  // (N,3) f32
    const float* rotation = (const float*)d_in[1];  // (N,4) f32
    float* out = (float*)d_out;                     // (N,3,3) f32
    const int N = in_sizes[0] / 3;
    const int grid = (N + IPB - 1) / IPB;
    gauss_cov_kernel<<<grid, TPB, 0, stream>>>(scaling, rotation, out, N);
}